// L3_Deep_6030134084107
// MI455X (gfx1250) — compile-verified
//
#include <hip/hip_runtime.h>
#include <math.h>

#define Nn 512
#define Dd 256
#define FDIM 32        // 31 features + 1 zero pad column (K multiple of 4 for WMMA)
#define JSWEEPS 10

typedef float v2f __attribute__((ext_vector_type(2)));
typedef float v8f __attribute__((ext_vector_type(8)));

// ---------------------------------------------------------------------------
// f32 WMMA GEMM, compile-time K/N strides: C[M,N] = A[M,K] @ B[K,N]
//   + (scale_ptr ? (1 + *scale_ptr) * addmat : 0)  + (bias ? bias[col] : 0)
//   optional ReLU.  M,N multiples of 16; K multiple of 16. One wave per tile,
//   K unrolled x4 -> 4 back-to-back V_WMMA_F32_16X16X4_F32 per iteration with
//   batched immediate-offset loads (waits spread instead of drained).
// Fragment layout per CDNA5 ISA (V_WMMA_F32_16X16X4_F32):
//   A (16x4): lane&15 = M row; lane>>4 selects K{0,1} vs K{2,3} across 2 VGPRs
//   B (4x16): lane&15 = N col; same K-half split
//   C (16x16): VGPR g -> M = g + 8*(lane>>4), N = lane&15
// ---------------------------------------------------------------------------
template <int KD, int ND>
__global__ __launch_bounds__(256)
void k_gemm_f32_t(const float* __restrict__ A, const float* __restrict__ B,
                  float* __restrict__ C, int Mdim,
                  const float* __restrict__ bias,
                  const float* __restrict__ addmat,
                  const float* __restrict__ scale_ptr,
                  int relu)
{
    constexpr int tilesN = ND >> 4;
    const int tiles = (Mdim >> 4) * tilesN;
    const int wave  = (int)((blockIdx.x * blockDim.x + threadIdx.x) >> 5);
    if (wave >= tiles) return;                 // wave-uniform: EXEC stays all-1s
    const int lane = threadIdx.x & 31;
    const int tm   = (wave / tilesN) << 4;
    const int tn   = (wave % tilesN) << 4;
    const int r    = lane & 15;
    const int half = lane >> 4;
    const int koff = half << 1;

    v8f acc = {0.f, 0.f, 0.f, 0.f, 0.f, 0.f, 0.f, 0.f};
    const float* Ap = A + (size_t)(tm + r) * KD + koff;     // lane's A stream
    const float* Bp = B + (size_t)koff * ND + tn + r;       // lane's B stream

    for (int k = 0; k < KD; k += 16) {
        v2f a0, a1, a2, a3, b0, b1, b2, b3;
        // A: four contiguous f32 pairs (b64 loads, immediate offsets)
        a0.x = Ap[k +  0]; a0.y = Ap[k +  1];
        a1.x = Ap[k +  4]; a1.y = Ap[k +  5];
        a2.x = Ap[k +  8]; a2.y = Ap[k +  9];
        a3.x = Ap[k + 12]; a3.y = Ap[k + 13];
        // B: eight b32 loads, compile-time row stride -> immediate offsets
        b0.x = Bp[(size_t)(k +  0) * ND]; b0.y = Bp[(size_t)(k +  1) * ND];
        b1.x = Bp[(size_t)(k +  4) * ND]; b1.y = Bp[(size_t)(k +  5) * ND];
        b2.x = Bp[(size_t)(k +  8) * ND]; b2.y = Bp[(size_t)(k +  9) * ND];
        b3.x = Bp[(size_t)(k + 12) * ND]; b3.y = Bp[(size_t)(k + 13) * ND];
        acc = __builtin_amdgcn_wmma_f32_16x16x4_f32(false, a0, false, b0,
                                                    (short)0, acc, false, false);
        acc = __builtin_amdgcn_wmma_f32_16x16x4_f32(false, a1, false, b1,
                                                    (short)0, acc, false, false);
        acc = __builtin_amdgcn_wmma_f32_16x16x4_f32(false, a2, false, b2,
                                                    (short)0, acc, false, false);
        acc = __builtin_amdgcn_wmma_f32_16x16x4_f32(false, a3, false, b3,
                                                    (short)0, acc, false, false);
    }

    float sc = 0.f;
    if (scale_ptr) sc = 1.0f + *scale_ptr;
#pragma unroll
    for (int g = 0; g < 8; ++g) {
        const int row = tm + g + (half << 3);
        const int col = tn + r;
        float v = acc[g];
        if (addmat) v += sc * addmat[(size_t)row * ND + col];
        if (bias)   v += bias[col];
        if (relu)   v = v > 0.f ? v : 0.f;
        C[(size_t)row * ND + col] = v;
    }
}

// ------------------------- prep: Af, degree, sum_j a_ij*a_ji -----------------
__global__ void k_prep(const int* __restrict__ a, float* __restrict__ Af,
                       float* __restrict__ deg, float* __restrict__ dd)
{
    __shared__ float s1[256], s2[256];
    const int i = blockIdx.x, t = threadIdx.x;
    float sd = 0.f, sdd = 0.f;
    for (int j = t; j < Nn; j += blockDim.x) {
        int v = a[i * Nn + j];
        Af[i * Nn + j] = (float)v;
        sd  += (float)v;
        sdd += (float)(v * a[j * Nn + i]);
    }
    s1[t] = sd; s2[t] = sdd; __syncthreads();
    for (int o = 128; o > 0; o >>= 1) {
        if (t < o) { s1[t] += s1[t + o]; s2[t] += s2[t + o]; }
        __syncthreads();
    }
    if (t == 0) { deg[i] = s1[0]; dd[i] = s2[0]; }
}

__global__ void k_buildP(const float* __restrict__ Af, const float* __restrict__ deg,
                         float* __restrict__ P)
{
    for (int idx = blockIdx.x * blockDim.x + threadIdx.x; idx < Nn * Nn;
         idx += gridDim.x * blockDim.x) {
        int i = idx >> 9;
        P[idx] = Af[idx] * (1.0f / (deg[i] + 1e-10f));
    }
}

__global__ void k_bits(const int* __restrict__ a, unsigned int* __restrict__ Ab)
{
    int idx = blockIdx.x * blockDim.x + threadIdx.x;
    if (idx >= Nn * (Nn / 32)) return;
    int i = idx / (Nn / 32), w = idx % (Nn / 32);
    unsigned int bits = 0u;
    for (int b = 0; b < 32; ++b)
        if (a[i * Nn + w * 32 + b]) bits |= (1u << b);
    Ab[idx] = bits;
}

__global__ void k_zero(float* __restrict__ p, int n)
{
    for (int i = blockIdx.x * blockDim.x + threadIdx.x; i < n;
         i += gridDim.x * blockDim.x) p[i] = 0.f;
}

// ------------------------- Jacobi eigensolver (persistent 1 WG) --------------
__device__ __forceinline__ void jac_pair(int rnd, int p, int& i, int& j)
{
    if (p == 0) { i = Nn - 1; j = rnd % (Nn - 1); }
    else { i = (rnd + p) % (Nn - 1); j = (rnd + (Nn - 1) - p) % (Nn - 1); }
}

__global__ void k_init_LV(const int* __restrict__ a, const float* __restrict__ deg,
                          float* __restrict__ Mj, float* __restrict__ Vj)
{
    for (int idx = blockIdx.x * blockDim.x + threadIdx.x; idx < Nn * Nn;
         idx += gridDim.x * blockDim.x) {
        int i = idx >> 9, j = idx & (Nn - 1);
        Mj[idx] = (i == j ? deg[i] : 0.f) - (float)a[idx];
        Vj[idx] = (i == j) ? 1.f : 0.f;
    }
}

__global__ __launch_bounds__(1024)
void k_jacobi(float* __restrict__ Mj, float* __restrict__ Vj, int sweeps)
{
    __shared__ float cs[Nn / 2], sn[Nn / 2];
    const int tid = threadIdx.x;
    for (int sw = 0; sw < sweeps; ++sw) {
        for (int rnd = 0; rnd < Nn - 1; ++rnd) {
            if (tid < Nn / 2) {                       // rotation angles
                int p, q; jac_pair(rnd, tid, p, q);
                float apq = Mj[(size_t)p * Nn + q];
                float c = 1.f, s = 0.f;
                if (fabsf(apq) > 1e-12f) {
                    float app = Mj[(size_t)p * Nn + p];
                    float aqq = Mj[(size_t)q * Nn + q];
                    float tau = (aqq - app) / (2.f * apq);
                    float t = (tau >= 0.f ? 1.f : -1.f) /
                              (fabsf(tau) + sqrtf(1.f + tau * tau));
                    c = 1.f / sqrtf(1.f + t * t);
                    s = t * c;
                }
                cs[tid] = c; sn[tid] = s;
            }
            __syncthreads();
            // column phase: M <- M*J, V <- V*J  (256 disjoint pairs)
            for (int u = tid; u < (Nn / 2) * Nn; u += blockDim.x) {
                int pr = u >> 9, i = u & (Nn - 1);
                int p, q; jac_pair(rnd, pr, p, q);
                float c = cs[pr], s = sn[pr];
                size_t ip = (size_t)i * Nn + p, iq = (size_t)i * Nn + q;
                float mp = Mj[ip], mq = Mj[iq];
                Mj[ip] = c * mp - s * mq; Mj[iq] = s * mp + c * mq;
                float vp = Vj[ip], vq = Vj[iq];
                Vj[ip] = c * vp - s * vq; Vj[iq] = s * vp + c * vq;
            }
            __syncthreads();
            // row phase: M <- J^T * M
            for (int u = tid; u < (Nn / 2) * Nn; u += blockDim.x) {
                int pr = u >> 9, j = u & (Nn - 1);
                int p, q; jac_pair(rnd, pr, p, q);
                float c = cs[pr], s = sn[pr];
                size_t pj = (size_t)p * Nn + j, qj = (size_t)q * Nn + j;
                float mp = Mj[pj], mq = Mj[qj];
                Mj[pj] = c * mp - s * mq; Mj[qj] = s * mp + c * mq;
            }
            __syncthreads();
        }
    }
}

__global__ void k_eig_select(const float* __restrict__ Mj,
                             const float* __restrict__ Vj, float* __restrict__ pe)
{
    __shared__ float lam[Nn];
    const int t = threadIdx.x;               // 512 threads
    lam[t] = Mj[(size_t)t * Nn + t];
    __syncthreads();
    float li = lam[t];
    int rank = 0;
    for (int j = 0; j < Nn; ++j) {
        float lj = lam[j];
        if (lj < li || (lj == li && j < t)) rank++;
    }
    if (rank >= 1 && rank <= 8) {
        int c = rank - 1;
        for (int row = 0; row < Nn; ++row)
            pe[row * 8 + c] = Vj[(size_t)row * Nn + t];
    }
}

__global__ void k_signfix(const float* __restrict__ pe, float* __restrict__ F)
{
    __shared__ int firstI;
    const int c = blockIdx.x, t = threadIdx.x;
    if (t == 0) firstI = Nn;
    __syncthreads();
    for (int r = t; r < Nn; r += blockDim.x)
        if (fabsf(pe[r * 8 + c]) > 1e-6f) atomicMin(&firstI, r);
    __syncthreads();
    float flip = 1.f;
    if (firstI < Nn && pe[firstI * 8 + c] < 0.f) flip = -1.f;
    for (int r = t; r < Nn; r += blockDim.x)
        F[r * FDIM + c] = pe[r * 8 + c] * flip;
}

// ------------------------- misc feature kernels ------------------------------
__global__ void k_diag(const float* __restrict__ Pk, float* __restrict__ F, int col)
{
    int i = threadIdx.x;                      // 512 threads
    F[i * FDIM + col] = Pk[(size_t)i * Nn + i];
}

#define MAXD 128
__global__ void k_k4(const int* __restrict__ a, const unsigned int* __restrict__ Ab,
                     float* __restrict__ k4)
{
    __shared__ int nbr[MAXD];
    __shared__ unsigned int rowv[Nn / 32];
    __shared__ int nd, cnt;
    const int v = blockIdx.x, t = threadIdx.x;
    if (t == 0) { nd = 0; cnt = 0; }
    __syncthreads();
    for (int j = t; j < Nn; j += blockDim.x)
        if (a[v * Nn + j]) { int p = atomicAdd(&nd, 1); if (p < MAXD) nbr[p] = j; }
    if (t < Nn / 32) rowv[t] = Ab[v * (Nn / 32) + t];
    __syncthreads();
    const int n = nd < MAXD ? nd : MAXD;
    int local = 0;
    for (int u = t; u < n * n; u += blockDim.x) {
        int i = u / n, j = u % n;
        if (i < j) {
            int bi = nbr[i], bj = nbr[j];
            if ((Ab[bi * (Nn / 32) + (bj >> 5)] >> (bj & 31)) & 1u) {
                const unsigned int* ri = Ab + bi * (Nn / 32);
                const unsigned int* rj = Ab + bj * (Nn / 32);
#pragma unroll
                for (int w = 0; w < Nn / 32; ++w)
                    local += __popc(ri[w] & rj[w] & rowv[w]);
            }
        }
    }
    atomicAdd(&cnt, local);
    __syncthreads();
    if (t == 0) k4[v] = (float)cnt * (1.0f / 3.0f);   // triangles in N(v)
}

__global__ void k_topk(const float* __restrict__ A4, float* __restrict__ F)
{
    __shared__ float vals[Nn];
    __shared__ float rv[128];
    __shared__ int   ri[128];
    const int row = blockIdx.x, t = threadIdx.x;      // 128 threads
    for (int j = t; j < Nn; j += 128) vals[j] = A4[(size_t)row * Nn + j];
    __syncthreads();
    for (int pass = 0; pass < 8; ++pass) {
        float best = -3.4e38f; int bi = Nn;
        for (int j = t; j < Nn; j += 128) {
            float x = vals[j];
            if (x > best || (x == best && j < bi)) { best = x; bi = j; }
        }
        rv[t] = best; ri[t] = bi; __syncthreads();
        for (int o = 64; o > 0; o >>= 1) {
            if (t < o) {
                if (rv[t + o] > rv[t] || (rv[t + o] == rv[t] && ri[t + o] < ri[t])) {
                    rv[t] = rv[t + o]; ri[t] = ri[t + o];
                }
            }
            __syncthreads();
        }
        if (t == 0) { F[row * FDIM + 20 + pass] = rv[0]; vals[ri[0]] = -3.4e38f; }
        __syncthreads();
    }
}

__global__ void k_cr(const float* __restrict__ Af, const float* __restrict__ A2,
                     const float* __restrict__ k4, const float* __restrict__ deg,
                     const float* __restrict__ dd, float* __restrict__ F)
{
    __shared__ float sE[256], sW[256], sT[256];
    const int i = blockIdx.x, t = threadIdx.x;
    float e = 0.f, w = 0.f, tt = 0.f;
    for (int j = t; j < Nn; j += 256) {
        float aij = Af[(size_t)i * Nn + j];
        float a2  = A2[(size_t)i * Nn + j];
        float m   = aij + (i == j ? 1.f : 0.f);
        float dm  = m * (a2 + aij);           // M ∘ (M@A),  M@A = A^2 + A
        e  += dm;
        w  += dm * (dm - 1.f);
        tt += a2 * aij;
    }
    sE[t] = e; sW[t] = w; sT[t] = tt; __syncthreads();
    for (int o = 128; o > 0; o >>= 1) {
        if (t < o) { sE[t] += sE[t + o]; sW[t] += sW[t + o]; sT[t] += sT[t + o]; }
        __syncthreads();
    }
    if (t == 0) {
        float s = deg[i] + 1.f;
        float E = 0.5f * sE[0], W = 0.5f * sW[0], tri = 0.5f * sT[0];
        float T  = tri + k4[i];
        float c3 = T;
        float c2 = W - 3.f * T;
        float c1 = E * (s - 2.f) - 2.f * c2 - 3.f * c3;
        float tot = s * (s - 1.f) * (s - 2.f) * (1.f / 6.f);
        float c0 = tot - c1 - c2 - c3;
        if (s < 3.f) { c0 = c1 = c2 = c3 = 0.f; }
        float den = 1.0f / (c0 + c1 + c2 + c3 + 1e-10f);
        F[i * FDIM + 16] = c0 * den; F[i * FDIM + 17] = c1 * den;
        F[i * FDIM + 18] = c2 * den; F[i * FDIM + 19] = c3 * den;
        F[i * FDIM + 28] = deg[i];
        F[i * FDIM + 29] = deg[i] * deg[i];
        F[i * FDIM + 30] = dd[i];
    }
}

__global__ void k_padEw(const float* __restrict__ e_w, float* __restrict__ Ewp)
{
    int idx = blockIdx.x * blockDim.x + threadIdx.x;  // 32*256
    if (idx >= FDIM * Dd) return;
    int row = idx / Dd;
    Ewp[idx] = (row < 31) ? e_w[idx] : 0.f;
}

__global__ void k_colsum(const float* __restrict__ x, float* __restrict__ out)
{
    const int c = threadIdx.x;                // 256 threads
    float s = 0.f;
    for (int r = 0; r < Nn; ++r) s += x[(size_t)r * Dd + c];
    out[c] = s;
}

// ---------------------------------------------------------------------------
static inline int gemm_blocks(int M_, int N_)
{
    int tiles = (M_ / 16) * (N_ / 16);
    return (tiles * 32 + 255) / 256;
}

extern "C" void kernel_launch(void* const* d_in, const int* in_sizes, int n_in,
                              void* d_out, int out_size, void* d_ws, size_t ws_size,
                              hipStream_t stream)
{
    const int*   a   = (const int*)  d_in[0];
    const float* e_w = (const float*)d_in[1];
    const float* e_b = (const float*)d_in[2];
    const float* w1  = (const float*)d_in[3];
    const float* b1  = (const float*)d_in[4];
    const float* w2  = (const float*)d_in[5];
    const float* b2  = (const float*)d_in[6];
    const float* eps = (const float*)d_in[7];
    float* out = (float*)d_out;

    // ---- carve workspace (~10.2 MB of floats) ----
    float* ws = (float*)d_ws;
    size_t off = 0;
    auto take = [&](size_t n) { float* p = ws + off; off += n; return p; };
    float* Af  = take((size_t)Nn * Nn);
    float* P   = take((size_t)Nn * Nn);
    float* Q1  = take((size_t)Nn * Nn);
    float* Q2  = take((size_t)Nn * Nn);
    float* A2  = take((size_t)Nn * Nn);
    float* A4  = take((size_t)Nn * Nn);
    float* Mj  = take((size_t)Nn * Nn);
    float* Vj  = take((size_t)Nn * Nn);
    float* xb  = take((size_t)Nn * Dd);
    float* agg = take((size_t)Nn * Dd);
    float* hb  = take((size_t)Nn * Dd);
    float* F   = take((size_t)Nn * FDIM);
    float* Ewp = take((size_t)FDIM * Dd);
    float* pe  = take((size_t)Nn * 8);
    float* deg = take(Nn);
    float* dd  = take(Nn);
    float* k4  = take(Nn);
    unsigned int* Ab = (unsigned int*)take(Nn * (Nn / 32));

    // ---- prep ----
    k_prep  <<<Nn, 256, 0, stream>>>(a, Af, deg, dd);
    k_bits  <<<(Nn * (Nn / 32) + 255) / 256, 256, 0, stream>>>(a, Ab);
    k_buildP<<<512, 256, 0, stream>>>(Af, deg, P);
    k_zero  <<<64, 256, 0, stream>>>(F, Nn * FDIM);

    // ---- Laplacian PE (Jacobi eigensolver) -> F[:,0:8] ----
    k_init_LV   <<<512, 256, 0, stream>>>(a, deg, Mj, Vj);
    k_jacobi    <<<1, 1024, 0, stream>>>(Mj, Vj, JSWEEPS);
    k_eig_select<<<1, 512, 0, stream>>>(Mj, Vj, pe);
    k_signfix   <<<8, 256, 0, stream>>>(pe, F);

    // ---- RW PE: diag(P^k), k=1..8 -> F[:,8:16] ----
    k_diag<<<1, 512, 0, stream>>>(P, F, 8);
    const float* cur = P;
    float* buf[2] = {Q1, Q2};
    int which = 0;
    for (int s = 1; s < 8; ++s) {
        float* dst = buf[which]; which ^= 1;
        k_gemm_f32_t<Nn, Nn><<<gemm_blocks(Nn, Nn), 256, 0, stream>>>(
            cur, P, dst, Nn, nullptr, nullptr, nullptr, 0);
        k_diag<<<1, 512, 0, stream>>>(dst, F, 8 + s);
        cur = dst;
    }

    // ---- A^2, A^4, top-8 -> F[:,20:28] ----
    k_gemm_f32_t<Nn, Nn><<<gemm_blocks(Nn, Nn), 256, 0, stream>>>(
        Af, Af, A2, Nn, nullptr, nullptr, nullptr, 0);
    k_gemm_f32_t<Nn, Nn><<<gemm_blocks(Nn, Nn), 256, 0, stream>>>(
        A2, A2, A4, Nn, nullptr, nullptr, nullptr, 0);
    k_topk<<<Nn, 128, 0, stream>>>(A4, F);

    // ---- K4 per vertex (bitset triangles) + CR + degree -> F[:,16:20],[28:31] ----
    k_k4<<<Nn, 128, 0, stream>>>(a, Ab, k4);
    k_cr<<<Nn, 256, 0, stream>>>(Af, A2, k4, deg, dd, F);

    // ---- embedding: x = F @ e_w + e_b ----
    k_padEw<<<(FDIM * Dd + 255) / 256, 256, 0, stream>>>(e_w, Ewp);
    k_gemm_f32_t<FDIM, Dd><<<gemm_blocks(Nn, Dd), 256, 0, stream>>>(
        F, Ewp, xb, Nn, e_b, nullptr, nullptr, 0);

    // ---- 3 GIN layers ----
    for (int l = 0; l < 3; ++l) {
        // agg = A@x + (1+eps[l])*x
        k_gemm_f32_t<Nn, Dd><<<gemm_blocks(Nn, Dd), 256, 0, stream>>>(
            Af, xb, agg, Nn, nullptr, xb, eps + l, 0);
        // h = relu(agg @ w1[l] + b1[l])
        k_gemm_f32_t<Dd, Dd><<<gemm_blocks(Nn, Dd), 256, 0, stream>>>(
            agg, w1 + (size_t)l * Dd * Dd, hb, Nn, b1 + l * Dd,
            nullptr, nullptr, 1);
        // x = h @ w2[l] + b2[l]
        k_gemm_f32_t<Dd, Dd><<<gemm_blocks(Nn, Dd), 256, 0, stream>>>(
            hb, w2 + (size_t)l * Dd * Dd, xb, Nn, b2 + l * Dd,
            nullptr, nullptr, 0);
    }

    // ---- out = x.sum(0) ----
    k_colsum<<<1, 256, 0, stream>>>(xb, out);
}